// NGCF_layer_81398220194344
// MI455X (gfx1250) — compile-verified
//
#include <hip/hip_runtime.h>

// ---------------------------------------------------------------------------
// NGCF layer for MI455X (gfx1250).
//
// Algebraic simplification of the reference:
//   r1[v] = feature[v] * s1[v],            s1[v] = sum_{e:dst=v} norm[e]
//   r2[v] = feature[v] (hadamard) g[v],    g[v]  = sum_{e:dst=v} feature[src[e]]*norm[e]
//   out   = ((s1+1)*feature) @ Wg^T + (feature.*g) @ Wie^T
//   out   = LeakyReLU(out, 0.2); out = out / max(||out||_2, 1e-12) rowwise
//
// Phase 1 (edge scatter): L2-resident (feature+g = 51 MB << 192 MB L2),
//   f32 global atomics. Phase 2 (node GEMM): f32 WMMA 16x16x4, one wave32
//   per 16-row tile, full-precision f32 path since GEMM cost (1.6 GFLOP)
//   is negligible vs. memory (~51 MB @ 23.3 TB/s).
// ---------------------------------------------------------------------------

typedef __attribute__((ext_vector_type(2))) float v2f;
typedef __attribute__((ext_vector_type(8))) float v8f;

#define NNODES 100000
#define NEDGES 1000000
#define DIM    64

// ---------------- workspace zero-init (required every call) ----------------
__global__ void zero_ws_kernel(float* __restrict__ p, int n) {
    int i = blockIdx.x * blockDim.x + threadIdx.x;
    if (i < n) p[i] = 0.0f;
}

// ---------------- phase 1: edge scatter (one wave32 per edge) --------------
// lane L owns columns {2L, 2L+1}; lane 0 additionally accumulates s1[dst].
__global__ __launch_bounds__(256) void edge_scatter_kernel(
    const float* __restrict__ feature,
    const float* __restrict__ in_d,
    const int*   __restrict__ src,
    const int*   __restrict__ dst,
    float* __restrict__ g,      // [N,64]
    float* __restrict__ s1)     // [N]
{
    int gid  = blockIdx.x * 256 + threadIdx.x;
    int e    = gid >> 5;
    int lane = gid & 31;
    if (e >= NEDGES) return;

    int s = src[e];
    int d = dst[e];
    // norm = 1/(sqrt(deg_s)*sqrt(deg_d)) = rsqrt(deg_s*deg_d)
    float norm = rsqrtf(in_d[s] * in_d[d]);

    const float* fs = feature + (long)s * DIM + 2 * lane;
    float fx = fs[0];
    float fy = fs[1];

    float* gd = g + (long)d * DIM + 2 * lane;
    unsafeAtomicAdd(gd + 0, fx * norm);   // -> global_atomic_add_f32 (no return)
    unsafeAtomicAdd(gd + 1, fy * norm);
    if (lane == 0) unsafeAtomicAdd(s1 + d, norm);
}

// ---------------- phase 2: fused scale + dual GEMM + epilogue --------------
// One wave32 per 16-row node tile (N = 6250*16 exactly, no tail).
// out[16,64] = A1[16,64] @ Wg^T + A2[16,64] @ Wie^T with
//   A1 = (s1+1)*feature, A2 = feature.*g
// as 4 N-tiles of 16x16, K in steps of 4 via V_WMMA_F32_16X16X4_F32.
//
// f32 WMMA fragment layout (ISA 7.12.2):
//   A 16x4:  lane<16: M=lane, v[0]=K0 v[1]=K1 ; lane>=16: M=lane-16, K2/K3
//   B 4x16:  mirrored (lane<16: N=lane, v[0]=K0 v[1]=K1 ; high half K2/K3)
//   C 16x16: v[r], lane<16 -> row r, lane>=16 -> row r+8, col = lane&15
__global__ __launch_bounds__(32) void node_gemm_kernel(
    const float* __restrict__ feature,
    const float* __restrict__ Wg,    // [64,64] row-major, out = x @ Wg^T
    const float* __restrict__ Wie,
    const float* __restrict__ g,
    const float* __restrict__ s1,
    float* __restrict__ out)
{
    const int tile = blockIdx.x;          // 0..6249
    const int row0 = tile * 16;
    const int lane = threadIdx.x;         // 0..31
    const int m    = lane & 15;           // M (and B's N) index for this lane
    const int kh   = (lane >> 4) * 2;     // K sub-offset: 0 (low half) / 2 (high)
    const int row  = row0 + m;

    v8f c[4] = {};                        // 4 N-tiles of the 16x64 output

    const float sc = s1[row] + 1.0f;      // (s1+1) scale for the W_gcn path
    const float* frow = feature + (long)row * DIM;
    const float* grow = g       + (long)row * DIM;

    for (int k0 = 0; k0 < DIM; k0 += 4) {
        const int ko = k0 + kh;
        v2f f2 = *(const v2f*)(frow + ko);
        v2f g2 = *(const v2f*)(grow + ko);
        v2f a1 = f2 * sc;                 // A-frag of (s1+1)*feature
        v2f a2 = f2 * g2;                 // A-frag of feature.*g

#pragma unroll
        for (int t = 0; t < 4; ++t) {
            const int wrow = t * 16 + m;  // W row == output column n
            v2f bg = *(const v2f*)(Wg  + wrow * DIM + ko);  // B[k][n] = W[n][k]
            v2f bi = *(const v2f*)(Wie + wrow * DIM + ko);
            c[t] = __builtin_amdgcn_wmma_f32_16x16x4_f32(
                       false, a1, false, bg, (short)0, c[t], false, false);
            c[t] = __builtin_amdgcn_wmma_f32_16x16x4_f32(
                       false, a2, false, bi, (short)0, c[t], false, false);
        }
    }

    // Epilogue: LeakyReLU(0.2) -> rowwise L2 normalize -> store.
    // Row of element c[t][r] is (r + 8*(lane>>4)); columns live on the 16
    // lanes of the same half-wave, so xor-masks 1,2,4,8 reduce exactly one row.
#pragma unroll
    for (int r = 0; r < 8; ++r) {
        float acc = 0.0f;
#pragma unroll
        for (int t = 0; t < 4; ++t) {
            float v = c[t][r];
            v = (v >= 0.0f) ? v : 0.2f * v;
            c[t][r] = v;
            acc += v * v;
        }
        acc += __shfl_xor(acc, 1);
        acc += __shfl_xor(acc, 2);
        acc += __shfl_xor(acc, 4);
        acc += __shfl_xor(acc, 8);
        const float scale = 1.0f / fmaxf(sqrtf(acc), 1e-12f);

        const int orow = row0 + r + (lane >> 4) * 8;
        float* orowp = out + (long)orow * DIM;
#pragma unroll
        for (int t = 0; t < 4; ++t)
            orowp[t * 16 + m] = c[t][r] * scale;
    }
}

// ---------------------------------------------------------------------------
extern "C" void kernel_launch(void* const* d_in, const int* in_sizes, int n_in,
                              void* d_out, int out_size, void* d_ws, size_t ws_size,
                              hipStream_t stream) {
    const float* feature = (const float*)d_in[0];   // [N,64]
    const float* in_d    = (const float*)d_in[1];   // [N,1]
    const float* Wg      = (const float*)d_in[2];   // [64,64]
    const float* Wie     = (const float*)d_in[3];   // [64,64]
    const int*   src     = (const int*)d_in[4];     // [E]
    const int*   dst     = (const int*)d_in[5];     // [E]
    float*       out     = (float*)d_out;           // [N,64]

    float* g  = (float*)d_ws;                       // [N,64] = 25.6 MB
    float* s1 = g + (size_t)NNODES * DIM;           // [N]

    const int zn = NNODES * (DIM + 1);
    zero_ws_kernel<<<(zn + 255) / 256, 256, 0, stream>>>(g, zn);

    // E waves, 8 edges per 256-thread block -> 125000 blocks
    edge_scatter_kernel<<<(NEDGES * 32) / 256, 256, 0, stream>>>(
        feature, in_d, src, dst, g, s1);

    // 6250 sixteen-row tiles, one wave32 each
    node_gemm_kernel<<<NNODES / 16, 32, 0, stream>>>(
        feature, Wg, Wie, g, s1, out);
}